// YOLOv4_83056077570914
// MI455X (gfx1250) — compile-verified
//
#include <hip/hip_runtime.h>
#include <stdint.h>

// ---------------- problem constants (from reference) ----------------
#define B_      32
#define A_      3
#define Hh      64
#define Ww      64
#define HW      4096            // 64*64
#define CH      255             // A*(5+80)
#define C5      85
#define NCAND   (HW * A_)       // 12288 candidates per image
#define KDET    300
#define NMS_IOU 0.7f
#define SCORE_T 0.05f
#define OBJ_T   0.5f

// decode tiling: each block pipelines TWO 64-cell tiles through double-buffered LDS
#define CTILE   64              // spatial cells per tile
#define LSTR2   68              // floats; 272B row stride = 64 data + 4 pad DWORDs (TDM pad)
#define NTPB    2               // tiles per block (pipeline depth)
#define THREADS 64              // 2 wave32 per block

// ---------------- output layout (floats, concatenated in return order) ----
#define DEC_ELEMS   (B_ * HW * A_ * C5)           // 33,423,360
#define OFF_BOXES   (DEC_ELEMS)                   // [B,300,4]
#define OFF_SCORES  (OFF_BOXES + B_ * KDET * 4)   // [B,300]
#define OFF_LABELS  (OFF_SCORES + B_ * KDET)      // [B,300] (as float)
#define OFF_KEEP    (OFF_LABELS + B_ * KDET)      // [B,300] (as float 0/1)

// ---------------- workspace layout (floats) ----------------
#define WS_SCORES   0                              // [B,NCAND] gated (-1 if invalid)
#define WS_BOXES    (B_ * NCAND)                   // [B,NCAND,4] xyxy clamped
#define WS_LABELS   (WS_BOXES + B_ * NCAND * 4)    // [B,NCAND]  (as float)

// ---------------- fast math helpers ----------------
__device__ __forceinline__ float fast_sigmoid(float x) {
    float e = __builtin_amdgcn_exp2f(-1.4426950408889634f * x);  // v_exp_f32
    return __builtin_amdgcn_rcpf(1.0f + e);                      // v_rcp_f32
}
__device__ __forceinline__ float fast_exp(float x) {
    return __builtin_amdgcn_exp2f(1.4426950408889634f * x);
}
__device__ __forceinline__ float clamp01(float x) {
    return fminf(fmaxf(x, 0.0f), 1.0f);
}

// ---------------- CDNA5 async global -> LDS (ASYNCcnt path) ----------------
typedef int v4i_gcc __attribute__((vector_size(16)));
typedef __attribute__((address_space(1))) v4i_gcc* g_v4i_p;   // global int4*
typedef __attribute__((address_space(3))) v4i_gcc* l_v4i_p;   // LDS int4*

__device__ __forceinline__ void async_g2l_b128(const float* g, float* l) {
#if __has_builtin(__builtin_amdgcn_global_load_async_to_lds_b128)
    __builtin_amdgcn_global_load_async_to_lds_b128(
        (g_v4i_p)g,          // global source (16B)
        (l_v4i_p)l,          // LDS destination (16B)
        0, 0);               // imm offset, cpol
#else
    unsigned lds_off = (unsigned)(unsigned long long)(uintptr_t)l;
    asm volatile("global_load_async_to_lds_b128 %0, %1, off"
                 :
                 : "v"(lds_off), "v"((unsigned long long)(uintptr_t)g)
                 : "memory");
#endif
}
__device__ __forceinline__ void wait_async0() {
#if __has_builtin(__builtin_amdgcn_s_wait_asynccnt)
    __builtin_amdgcn_s_wait_asynccnt(0);
#else
    asm volatile("s_wait_asynccnt 0x0" ::: "memory");
#endif
}

// ---------------- CDNA5 Tensor Data Mover (TENSORcnt path) ----------------
#if __has_builtin(__builtin_amdgcn_tensor_load_to_lds)
#define USE_TDM 1
typedef unsigned tdm_v4u __attribute__((ext_vector_type(4)));
typedef int      tdm_v8i __attribute__((ext_vector_type(8)));
typedef int      tdm_v4i __attribute__((ext_vector_type(4)));

// Issue (no wait) a TDM load of a [85 rows x 64 floats] tile (row stride 4096
// floats in memory) into LDS, padding 4 DWORDs after every 64 DWORDs so the
// LDS row stride is 68 floats.
__device__ __forceinline__ void tdm_issue_tile(const float* gsrc, float* lds_dst) {
    unsigned lds_addr = (unsigned)(uintptr_t)
        ((__attribute__((address_space(3))) float*)lds_dst);
    unsigned long long ga = (unsigned long long)(uintptr_t)gsrc;

    tdm_v4u g0;
    g0.x = 1u;                                            // count=1, user descriptor
    g0.y = lds_addr;                                      // lds_addr
    g0.z = (unsigned)(ga & 0xFFFFFFFFull);                // global_addr[31:0]
    g0.w = (unsigned)((ga >> 32) & 0x01FFFFFFull)         // global_addr[56:32]
         | (2u << 30);                                    // type = 2 ("image")

    tdm_v8i g1;
    g1.s0 = (2 << 16)       // data_size = 4B
          | (1 << 20)       // pad_enable
          | (5 << 22)       // pad_interval: 2^(5+1)=64 DWORDs
          | (3 << 25);      // pad_amount: 3+1 = 4 DWORDs
    g1.s1 = (int)(4096u << 16);   // tensor_dim0[15:0] @ bits63:48 (=4096)
    g1.s2 = (int)(85u << 16);     // tensor_dim0 hi=0; tensor_dim1[15:0] @ bits95:80 (=85)
    g1.s3 = (int)((unsigned)CTILE << 16); // tensor_dim1 hi=0; tile_dim0 @ bits127:112 (=64)
    g1.s4 = 85;                   // tile_dim1 = 85; tile_dim2 = 0
    g1.s5 = 4096;                 // tensor_dim0_stride[31:0] = 4096 elements
    g1.s6 = 0;                    // stride hi / tensor_dim1_stride (unused, 2D)
    g1.s7 = 0;

    tdm_v4i g2 = {0, 0, 0, 0};                 // dims 2/3 unused (2D tile)
    tdm_v4i g3 = {0, 0, 0, 0};
    tdm_v8i g4 = {0, 0, 0, 0, 0, 0, 0, 0};     // extra payload, unused for 2D

    __builtin_amdgcn_tensor_load_to_lds(g0, g1, g2, g3, g4, 0);
}
#endif

// ---------------- per-tile decode compute (uniform control flow) ----------
__device__ __forceinline__ void process_tile(
    float* sm, int t, int b, int a, int toff,
    float* __restrict__ decoded,
    float* __restrict__ ws_scores,
    float* __restrict__ ws_boxes,
    float* __restrict__ ws_labels)
{
    const int   p   = toff + t;
    const int   xg  = p & (Ww - 1);
    const int   yg  = p >> 6;
    const float inv = 1.0f / 64.0f;
    // anchors / stride / grid:  [12,19,40]/8/64 , [16,36,28]/8/64
    const float AW[3] = {12.0f / 512.0f, 19.0f / 512.0f, 40.0f / 512.0f};
    const float AH[3] = {16.0f / 512.0f, 36.0f / 512.0f, 28.0f / 512.0f};

    float bx  = (fast_sigmoid(sm[0 * LSTR2 + t]) + (float)xg) * inv;
    float by  = (fast_sigmoid(sm[1 * LSTR2 + t]) + (float)yg) * inv;
    float bw  = fast_exp(sm[2 * LSTR2 + t]) * AW[a];
    float bh  = fast_exp(sm[3 * LSTR2 + t]) * AH[a];
    float obj = fast_sigmoid(sm[4 * LSTR2 + t]);
    sm[0 * LSTR2 + t] = bx;
    sm[1 * LSTR2 + t] = by;
    sm[2 * LSTR2 + t] = bw;
    sm[3 * LSTR2 + t] = bh;
    sm[4 * LSTR2 + t] = obj;

    float cmax = -1.0f;
    int   lab  = 0;
    #pragma unroll 4
    for (int c = 5; c < C5; ++c) {
        float s = fast_sigmoid(sm[c * LSTR2 + t]);
        sm[c * LSTR2 + t] = s;
        if (s > cmax) { cmax = s; lab = c - 5; }   // first-max, matches argmax
    }

    // xywh -> xyxy, clamp, gate
    const float tlx = bx - 0.5f * bw, tly = by - 0.5f * bh;
    const float x1 = clamp01(tlx),       y1 = clamp01(tly);
    const float x2 = clamp01(tlx + bw),  y2 = clamp01(tly + bh);
    const float score = cmax * obj;
    const float gated = (obj >= OBJ_T && score >= SCORE_T) ? score : -1.0f;

    const size_t bi = (size_t)b * NCAND + (size_t)p * A_ + a;
    ws_scores[bi] = gated;
    ws_labels[bi] = (float)lab;
    float4 bv; bv.x = x1; bv.y = y1; bv.z = x2; bv.w = y2;
    ((float4*)ws_boxes)[bi] = bv;

    __syncthreads();   // pass1 in-place writes visible before transposed reads

    // coalesced transposed write of decoded rows
    const size_t obase = ((size_t)b * HW + toff) * CH + (size_t)a * C5;
    for (int e = t; e < CTILE * C5; e += THREADS) {
        const int ce = e / C5;
        const int ch = e % C5;
        decoded[obase + (size_t)ce * CH + ch] = sm[ch * LSTR2 + ce];
    }
}

// ======================= decode kernel =======================
// grid: B * A * (HW / (2*CTILE)) blocks, 64 threads each; 2 tiles pipelined.
__global__ __launch_bounds__(THREADS) void decode_kernel(
    const float* __restrict__ in,          // [B, 255, 64, 64]
    float* __restrict__ decoded,           // [B, HW, A, 85] (== [B,HW,255])
    float* __restrict__ ws_scores,
    float* __restrict__ ws_boxes,
    float* __restrict__ ws_labels)
{
    __shared__ float sm[NTPB][C5 * LSTR2];   // 2 x 23,120 B = 46,240 B

    const int t    = threadIdx.x;
    const int blk  = blockIdx.x;
    const int TPB  = HW / (NTPB * CTILE);    // 32 tile-pairs per (b,a) plane
    const int b    = blk / (A_ * TPB);
    const int rem  = blk % (A_ * TPB);
    const int a    = rem / TPB;
    const int toff = (rem % TPB) * (NTPB * CTILE);

    const float* src = in + ((size_t)(b * CH + a * C5)) * HW + toff;

#ifdef USE_TDM
    // wave 0 issues both descriptor DMAs back-to-back (TENSORcnt -> 2)
    if (t < 32) {
        tdm_issue_tile(src,         sm[0]);
        tdm_issue_tile(src + CTILE, sm[1]);
        __builtin_amdgcn_s_wait_tensorcnt(1);   // tile0 landed (in-order per wave)
    }
    __syncthreads();
    process_tile(sm[0], t, b, a, toff, decoded, ws_scores, ws_boxes, ws_labels);

    if (t < 32) __builtin_amdgcn_s_wait_tensorcnt(0);  // tile1 landed
    __syncthreads();
    process_tile(sm[1], t, b, a, toff + CTILE, decoded, ws_scores, ws_boxes, ws_labels);
#else
    // fallback: per-lane async copies for both tiles, then process
    const int NCHUNK = C5 * (CTILE / 4);     // 1360 16-byte chunks per tile
    for (int tile = 0; tile < NTPB; ++tile) {
        for (int k = t; k < NCHUNK; k += THREADS) {
            const int c = k >> 4;            // CTILE/4 == 16 chunks per channel row
            const int j = k & 15;
            async_g2l_b128(src + (size_t)tile * CTILE + (size_t)c * HW + j * 4,
                           &sm[tile][c * LSTR2 + j * 4]);
        }
    }
    wait_async0();
    __syncthreads();
    process_tile(sm[0], t, b, a, toff, decoded, ws_scores, ws_boxes, ws_labels);
    __syncthreads();
    process_tile(sm[1], t, b, a, toff + CTILE, decoded, ws_scores, ws_boxes, ws_labels);
#endif
}

// ======================= top-k + NMS kernel =======================
// grid: B blocks, 512 threads (16 wave32) each.
__global__ __launch_bounds__(512) void topk_nms_kernel(
    const float* __restrict__ ws_scores,
    const float* __restrict__ ws_boxes,
    const float* __restrict__ ws_labels,
    float* __restrict__ out)
{
    __shared__ float sc[NCAND];            // 48 KB
    __shared__ float rs[512];
    __shared__ int   ri[512];
    __shared__ float tbx[KDET], tby[KDET], tbX[KDET], tbY[KDET];
    __shared__ float tsc[KDET], tlab[KDET], tarea[KDET];
    __shared__ int   tidx[KDET], keep[KDET];

    const int    t    = threadIdx.x;
    const int    b    = blockIdx.x;
    const size_t base = (size_t)b * NCAND;

    // ---- async bulk copy of this image's gated scores into LDS ----
    for (int k = t; k < NCAND / 4; k += 512)   // 3072 16-byte chunks
        async_g2l_b128(ws_scores + base + (size_t)k * 4, &sc[k * 4]);
    wait_async0();
    __syncthreads();

    // ---- iterative top-300 selection (stable: ties -> lowest index) ----
    for (int it = 0; it < KDET; ++it) {
        float bs = -1e30f;
        int   bi = NCAND;
        for (int j = t; j < NCAND; j += 512) {
            float v = sc[j];
            if (v > bs) { bs = v; bi = j; }        // ascending j => first max kept
        }
        rs[t] = bs; ri[t] = bi;
        __syncthreads();
        for (int s = 256; s > 0; s >>= 1) {
            if (t < s) {
                float v = rs[t + s]; int j = ri[t + s];
                if (v > rs[t] || (v == rs[t] && j < ri[t])) { rs[t] = v; ri[t] = j; }
            }
            __syncthreads();
        }
        if (t == 0) {
            int j = ri[0];
            tsc[it]  = rs[0];
            tidx[it] = j;
            sc[j]    = -1e30f;                      // remove from candidate set
        }
        __syncthreads();
    }

    // ---- gather candidate boxes/labels, init keep ----
    if (t < KDET) {
        const int    j  = tidx[t];
        const float* bp = ws_boxes + (base + (size_t)j) * 4;
        const float x1 = bp[0], y1 = bp[1], x2 = bp[2], y2 = bp[3];
        tbx[t] = x1; tby[t] = y1; tbX[t] = x2; tbY[t] = y2;
        tlab[t]  = ws_labels[base + j];
        keep[t]  = (tsc[t] > 0.0f) ? 1 : 0;         // invalid candidates have -1
        tarea[t] = fmaxf(x2 - x1, 0.0f) * fmaxf(y2 - y1, 0.0f);
    }
    __syncthreads();

    // ---- sequential NMS, parallel over j (mirrors reference fori_loop) ----
    for (int i = 0; i < KDET; ++i) {
        if (t > i && t < KDET && keep[i]) {
            const float ix1 = fmaxf(tbx[i], tbx[t]);
            const float iy1 = fmaxf(tby[i], tby[t]);
            const float ix2 = fminf(tbX[i], tbX[t]);
            const float iy2 = fminf(tbY[i], tbY[t]);
            const float inter = fmaxf(ix2 - ix1, 0.0f) * fmaxf(iy2 - iy1, 0.0f);
            const float iou   = inter / (tarea[i] + tarea[t] - inter + 1e-9f);
            if (iou > NMS_IOU) keep[t] = 0;
        }
        __syncthreads();
    }

    // ---- emit outputs ----
    float* ob = out + OFF_BOXES;
    float* os = out + OFF_SCORES;
    float* ol = out + OFF_LABELS;
    float* ok = out + OFF_KEEP;
    for (int k = t; k < KDET; k += 512) {
        const size_t o = (size_t)b * KDET + k;
        ob[o * 4 + 0] = tbx[k];
        ob[o * 4 + 1] = tby[k];
        ob[o * 4 + 2] = tbX[k];
        ob[o * 4 + 3] = tbY[k];
        os[o] = tsc[k];
        ol[o] = tlab[k];
        ok[o] = keep[k] ? 1.0f : 0.0f;
    }
}

// ======================= launcher =======================
extern "C" void kernel_launch(void* const* d_in, const int* in_sizes, int n_in,
                              void* d_out, int out_size, void* d_ws, size_t ws_size,
                              hipStream_t stream) {
    (void)in_sizes; (void)n_in; (void)out_size; (void)ws_size;
    const float* in  = (const float*)d_in[0];
    float*       out = (float*)d_out;
    float*       wsf = (float*)d_ws;

    float* ws_scores = wsf + WS_SCORES;
    float* ws_boxes  = wsf + WS_BOXES;
    float* ws_labels = wsf + WS_LABELS;

    decode_kernel<<<B_ * A_ * (HW / (NTPB * CTILE)), THREADS, 0, stream>>>(
        in, out, ws_scores, ws_boxes, ws_labels);
    topk_nms_kernel<<<B_, 512, 0, stream>>>(ws_scores, ws_boxes, ws_labels, out);
}